// data_share_block_24472723653316
// MI455X (gfx1250) — compile-verified
//
#include <hip/hip_runtime.h>

typedef float v4f __attribute__((ext_vector_type(4)));
typedef int   v4i __attribute__((ext_vector_type(4)));

#define ALPHA 0.5f
#define BETA  0.25f

constexpr int  NC    = 32;
constexpr int  NT    = 64;
constexpr int  NX    = 256;
constexpr int  NY    = 256;
constexpr long PLANE = (long)NX * NY;     // 65536 elements per t-plane
constexpr long CHSTR = (long)NT * PLANE;  // 4194304 elements per channel

// One thread handles one float4 at (t, x, y..y+3) for ALL 32 channels.
// Mask logic is computed once per thread and amortized over the channel loop.
__global__ __launch_bounds__(256) void masked_tshift_kernel(
    const float* __restrict__ k,
    const int*   __restrict__ mask,
    float*       __restrict__ out)
{
    const int idx = blockIdx.x * 256 + threadIdx.x;   // 0 .. NT*NX*NY/4 - 1
    const int y   = (idx & 63) << 2;                  // 0..252 step 4
    const int x   = (idx >> 6) & (NX - 1);            // 0..255
    const int t   = idx >> 14;                        // 0..63  (blocks never straddle t)

    const long moff = (long)x * NY + y;               // offset within a t-plane
    const long off  = (long)t * PLANE + moff;         // offset within a channel

    // Mask planes: pad with zeros at t boundaries exactly like the reference.
    v4i mc = *(const v4i*)(mask + (long)t * PLANE + moff);
    v4i mp = (v4i)(0);
    v4i mn = (v4i)(0);
    if (t > 0)      mp = *(const v4i*)(mask + (long)(t - 1) * PLANE + moff);
    if (t < NT - 1) mn = *(const v4i*)(mask + (long)(t + 1) * PLANE + moff);

    // Per-element blend coefficients. At most one of c1/c2/c3 is nonzero,
    // so the FMA blend below reproduces the reference bit-exactly.
    v4f c1, c2, c3;
#pragma unroll
    for (int i = 0; i < 4; ++i) {
        const bool bc = (mc[i] != 0);
        const bool bp = (mp[i] != 0);
        const bool bn = (mn[i] != 0);
        c2[i] = bc ? 1.0f : 0.0f;                    // m_cur
        c1[i] = (!bc && bp) ? ALPHA : 0.0f;          // m1 = ~m_cur & m_prev
        c3[i] = (bn && !bp && !bc) ? BETA : 0.0f;    // m3 = m_next & ~(m_prev|m_cur)
    }

    // Clamp neighbor offsets at the t boundaries; the matching coefficient is
    // provably zero there, so the clamped (duplicate) load value is never used.
    const long offP = (t > 0)      ? off - PLANE : off;
    const long offN = (t < NT - 1) ? off + PLANE : off;

#pragma unroll 4
    for (int c = 0; c < NC; ++c) {
        const float* kc = k + (long)c * CHSTR;

        // gfx1250 prefetch of next channel's center line (16 MB away, fresh line).
        if (c + 1 < NC) __builtin_prefetch(kc + CHSTR + off, 0, 3);

        v4f vp = *(const v4f*)(kc + offP);
        v4f vc = *(const v4f*)(kc + off);
        v4f vn = *(const v4f*)(kc + offN);

        v4f o;
#pragma unroll
        for (int i = 0; i < 4; ++i)
            o[i] = __builtin_fmaf(c1[i], vp[i],
                   __builtin_fmaf(c3[i], vn[i], c2[i] * vc[i]));

        // Output is never re-read: stream it past L2 (TH=NT) so the 512 MB
        // write stream doesn't evict k lines whose t+/-1 reuse we want in L2.
        __builtin_nontemporal_store(o, (v4f*)(out + (long)c * CHSTR + off));
    }
}

extern "C" void kernel_launch(void* const* d_in, const int* in_sizes, int n_in,
                              void* d_out, int out_size, void* d_ws, size_t ws_size,
                              hipStream_t stream) {
    const float* k    = (const float*)d_in[0];
    const int*   mask = (const int*)d_in[1];
    float*       out  = (float*)d_out;

    const int total4 = NT * NX * NY / 4;   // 1,048,576 float4 per channel-plane set
    dim3 grid(total4 / 256);               // 4096 blocks
    dim3 block(256);                       // 8 wave32 per block
    masked_tshift_kernel<<<grid, block, 0, stream>>>(k, mask, out);
}